// GNN_27152783245352
// MI455X (gfx1250) — compile-verified
//
#include <hip/hip_runtime.h>
#include <hip/hip_bf16.h>

#define N_NODES 50000
#define N_EDGES 800000
#define F_DIM   256           // F_IN == H == 256
#define N_CLS   40
#define ALPHA   0.5f

typedef __attribute__((ext_vector_type(2))) float v2f;
typedef __attribute__((ext_vector_type(8))) float v8f;

// ---------------- degree / edge weight prep ----------------

__global__ __launch_bounds__(256) void degrees_kernel(
    const long long* __restrict__ row, const long long* __restrict__ col,
    float* __restrict__ degO, float* __restrict__ degI)
{
    int e = blockIdx.x * 256 + threadIdx.x;
    if (e < N_EDGES) {
        atomicAdd(&degO[(int)row[e]], 1.0f);
        atomicAdd(&degI[(int)col[e]], 1.0f);
    }
}

__global__ __launch_bounds__(256) void inv_sqrt_kernel(float* __restrict__ d, int n)
{
    int i = blockIdx.x * 256 + threadIdx.x;
    if (i < n) {
        float v = d[i];
        d[i] = (v > 0.0f) ? rsqrtf(fmaxf(v, 1e-12f)) : 0.0f;
    }
}

__global__ __launch_bounds__(256) void edge_w_kernel(
    const long long* __restrict__ row, const long long* __restrict__ col,
    const float* __restrict__ invO, const float* __restrict__ invI,
    float* __restrict__ ew)
{
    int e = blockIdx.x * 256 + threadIdx.x;
    if (e < N_EDGES) ew[e] = invO[(int)row[e]] * invI[(int)col[e]];
}

// ---------------- sparse aggregation (both directions in one pass) ----------------
// one wave per edge; 32 lanes x 2 float4 = 256 features
__global__ __launch_bounds__(256) void scatter_kernel(
    const float* __restrict__ h,
    const long long* __restrict__ row, const long long* __restrict__ col,
    const float* __restrict__ ew,
    float* __restrict__ aggS, float* __restrict__ aggD)
{
    int e = blockIdx.x * 8 + (threadIdx.x >> 5);
    if (e >= N_EDGES) return;
    int lane = threadIdx.x & 31;
    int r = (int)row[e];
    int c = (int)col[e];
    float w = ew[e];
    const float4* __restrict__ hc = (const float4*)(h + (size_t)c * F_DIM);
    const float4* __restrict__ hr = (const float4*)(h + (size_t)r * F_DIM);
    float* __restrict__ ar = aggS + (size_t)r * F_DIM;   // adj_norm @ h   -> out[row]
    float* __restrict__ ac = aggD + (size_t)c * F_DIM;   // adj_t_norm @ h -> out[col]
#pragma unroll
    for (int t = 0; t < 2; ++t) {
        int q = lane + t * 32;          // float4 index within row
        float4 vc = hc[q];
        float4 vr = hr[q];
        int f = q * 4;
        atomicAdd(ar + f + 0, w * vc.x);
        atomicAdd(ar + f + 1, w * vc.y);
        atomicAdd(ar + f + 2, w * vc.z);
        atomicAdd(ar + f + 3, w * vc.w);
        atomicAdd(ac + f + 0, w * vr.x);
        atomicAdd(ac + f + 1, w * vr.y);
        atomicAdd(ac + f + 2, w * vr.z);
        atomicAdd(ac + f + 3, w * vr.w);
    }
}

// ---------------- fused dual GEMM: out = a*(As@Ws^T + bs) + (1-a)*(Ad@Wd^T + bd) ----
// K = 256 fixed. Block = 128 rows x 32 cols; 8 waves, each wave computes a 16x32 tile
// (two 16x16 WMMA column sub-tiles, so every A fragment feeds 2 WMMAs).
// Weight tiles live in LDS in K-major layout with an XOR-16 column swizzle:
//   element (k, j) stored at  k*32 + (j ^ ((((k>>1)&1)) << 4))
// Since the K loop steps by 4 and a lane reads kk = k + 2*lhalf, the swizzle bit
// equals lhalf, making the two half-waves hit disjoint 16-bank sets (conflict-free).
__global__ __launch_bounds__(256) void gemm_dir_kernel(
    const float* __restrict__ As, const float* __restrict__ Ad,
    const float* __restrict__ Ws, const float* __restrict__ Wd,
    const float* __restrict__ bs, const float* __restrict__ bd,
    float* __restrict__ Out, int M, int Fout, int doRelu)
{
    const int K = 256;
    __shared__ float sWs[K * 32];   // 32 KB
    __shared__ float sWd[K * 32];   // 32 KB  (total 64 KB)

    const int j0 = blockIdx.x * 32;
    const int m0 = blockIdx.y * 128;
    const int tid = threadIdx.x;

    // stage weight tiles: lanes carry the 32 output columns, float4 along K from
    // global (L2-resident), scalar conflict-free swizzled stores into LDS.
    for (int i = tid; i < 2048; i += 256) {
        int j  = i & 31;                 // column within tile
        int k4 = (i >> 5) << 2;          // 0,4,...,252
        int jr = j0 + j;
        int jrc = (jr < Fout) ? jr : 0;  // clamp; stores guarded later
        float4 w = *(const float4*)(Ws + (size_t)jrc * K + k4);
        float4 v = *(const float4*)(Wd + (size_t)jrc * K + k4);
        float wq[4] = {w.x, w.y, w.z, w.w};
        float vq[4] = {v.x, v.y, v.z, v.w};
#pragma unroll
        for (int q = 0; q < 4; ++q) {
            int k  = k4 + q;
            int js = j ^ (((k >> 1) & 1) << 4);
            sWs[k * 32 + js] = wq[q];
            sWd[k * 32 + js] = vq[q];
        }
    }
    __syncthreads();

    const int wave  = tid >> 5;
    const int lane  = tid & 31;
    const int l16   = lane & 15;
    const int lhalf = lane >> 4;
    const int mrow  = m0 + wave * 16;

    int arow = mrow + l16;
    if (arow >= M) arow = 0;                      // clamp; stores guarded below
    const float* __restrict__ pAs = As + (size_t)arow * K + 2 * lhalf;
    const float* __restrict__ pAd = Ad + (size_t)arow * K + 2 * lhalf;

    // swizzled per-lane column indices for the two 16-wide sub-tiles
    const int jsw0 = (l16)      ^ (lhalf << 4);
    const int jsw1 = (16 + l16) ^ (lhalf << 4);
    const int kbase = 64 * lhalf;                 // (2*lhalf)*32

    v8f accS0 = {0.f,0.f,0.f,0.f,0.f,0.f,0.f,0.f};
    v8f accS1 = {0.f,0.f,0.f,0.f,0.f,0.f,0.f,0.f};
    v8f accD0 = {0.f,0.f,0.f,0.f,0.f,0.f,0.f,0.f};
    v8f accD1 = {0.f,0.f,0.f,0.f,0.f,0.f,0.f,0.f};

#pragma unroll 4
    for (int k = 0; k < K; k += 4) {
        v2f aS = *(const v2f*)(pAs + k);          // A frag (global, b64)
        v2f aD = *(const v2f*)(pAd + k);
        int b = k * 32 + kbase;
        v2f bS0 = { sWs[b + jsw0], sWs[b + 32 + jsw0] };
        v2f bS1 = { sWs[b + jsw1], sWs[b + 32 + jsw1] };
        v2f bD0 = { sWd[b + jsw0], sWd[b + 32 + jsw0] };
        v2f bD1 = { sWd[b + jsw1], sWd[b + 32 + jsw1] };
        accS0 = __builtin_amdgcn_wmma_f32_16x16x4_f32(false, aS, false, bS0, (short)0, accS0, false, false);
        accS1 = __builtin_amdgcn_wmma_f32_16x16x4_f32(false, aS, false, bS1, (short)0, accS1, false, false);
        accD0 = __builtin_amdgcn_wmma_f32_16x16x4_f32(false, aD, false, bD0, (short)0, accD0, false, false);
        accD1 = __builtin_amdgcn_wmma_f32_16x16x4_f32(false, aD, false, bD1, (short)0, accD1, false, false);
    }

    const int ja = j0 + l16;
    const int jb = j0 + 16 + l16;
    const float bsa = (ja < Fout) ? bs[ja] : 0.0f;
    const float bda = (ja < Fout) ? bd[ja] : 0.0f;
    const float bsb = (jb < Fout) ? bs[jb] : 0.0f;
    const float bdb = (jb < Fout) ? bd[jb] : 0.0f;
#pragma unroll
    for (int r = 0; r < 8; ++r) {
        int m = mrow + r + 8 * lhalf;             // C/D layout: VGPR r -> M=r / M=8+r
        float va = ALPHA * (accS0[r] + bsa) + (1.0f - ALPHA) * (accD0[r] + bda);
        float vb = ALPHA * (accS1[r] + bsb) + (1.0f - ALPHA) * (accD1[r] + bdb);
        if (doRelu) { va = fmaxf(va, 0.0f); vb = fmaxf(vb, 0.0f); }
        if (m < M) {
            if (ja < Fout) Out[(size_t)m * Fout + ja] = va;
            if (jb < Fout) Out[(size_t)m * Fout + jb] = vb;
        }
    }
}

// ---------------- log_softmax over C=40, one wave per row, in place ----------------
__global__ __launch_bounds__(256) void logsoftmax_kernel(float* __restrict__ out, int M, int C)
{
    int e = blockIdx.x * 8 + (threadIdx.x >> 5);
    if (e >= M) return;
    int lane = threadIdx.x & 31;
    float* r = out + (size_t)e * C;
    float v0 = (lane < C) ? r[lane] : -3.4e38f;
    float v1 = (lane + 32 < C) ? r[lane + 32] : -3.4e38f;
    float mx = fmaxf(v0, v1);
#pragma unroll
    for (int o = 16; o > 0; o >>= 1) mx = fmaxf(mx, __shfl_xor(mx, o, 32));
    float s = ((lane < C) ? __expf(v0 - mx) : 0.0f) +
              ((lane + 32 < C) ? __expf(v1 - mx) : 0.0f);
#pragma unroll
    for (int o = 16; o > 0; o >>= 1) s += __shfl_xor(s, o, 32);
    float lse = mx + logf(s);
    if (lane < C) r[lane] = v0 - lse;
    if (lane + 32 < C) r[lane + 32] = v1 - lse;
}

// ---------------- launch ----------------

extern "C" void kernel_launch(void* const* d_in, const int* in_sizes, int n_in,
                              void* d_out, int out_size, void* d_ws, size_t ws_size,
                              hipStream_t stream)
{
    (void)in_sizes; (void)n_in; (void)out_size; (void)ws_size;

    const float*     x   = (const float*)d_in[0];
    const long long* ei  = (const long long*)d_in[1];
    const float* w_s1 = (const float*)d_in[2];
    const float* b_s1 = (const float*)d_in[3];
    const float* w_d1 = (const float*)d_in[4];
    const float* b_d1 = (const float*)d_in[5];
    const float* w_s2 = (const float*)d_in[6];
    const float* b_s2 = (const float*)d_in[7];
    const float* w_d2 = (const float*)d_in[8];
    const float* b_d2 = (const float*)d_in[9];
    const float* w_s3 = (const float*)d_in[10];
    const float* b_s3 = (const float*)d_in[11];
    const float* w_d3 = (const float*)d_in[12];
    const float* b_d3 = (const float*)d_in[13];
    float* out = (float*)d_out;

    const long long* row = ei;
    const long long* col = ei + N_EDGES;

    float* wsf = (float*)d_ws;
    size_t off = 0;
    float* invO = wsf + off; off += N_NODES;
    float* invI = wsf + off; off += N_NODES;
    float* ew   = wsf + off; off += N_EDGES;
    float* aggS = wsf + off; off += (size_t)N_NODES * F_DIM;
    float* aggD = wsf + off; off += (size_t)N_NODES * F_DIM;
    float* h1   = wsf + off; off += (size_t)N_NODES * F_DIM;
    float* h2   = wsf + off; off += (size_t)N_NODES * F_DIM;

    const int gE = (N_EDGES + 255) / 256;
    const int gScatter = (N_EDGES + 7) / 8;
    dim3 gGemmH((F_DIM + 31) / 32, (N_NODES + 127) / 128);
    dim3 gGemmC((N_CLS + 31) / 32, (N_NODES + 127) / 128);

    // degree -> inverse-sqrt -> per-edge weight
    hipMemsetAsync(invO, 0, (size_t)2 * N_NODES * sizeof(float), stream);
    degrees_kernel<<<gE, 256, 0, stream>>>(row, col, invO, invI);
    inv_sqrt_kernel<<<(2 * N_NODES + 255) / 256, 256, 0, stream>>>(invO, 2 * N_NODES);
    edge_w_kernel<<<gE, 256, 0, stream>>>(row, col, invO, invI, ew);

    // layer 1
    hipMemsetAsync(aggS, 0, (size_t)2 * N_NODES * F_DIM * sizeof(float), stream);
    scatter_kernel<<<gScatter, 256, 0, stream>>>(x, row, col, ew, aggS, aggD);
    gemm_dir_kernel<<<gGemmH, 256, 0, stream>>>(aggS, aggD, w_s1, w_d1, b_s1, b_d1,
                                                h1, N_NODES, F_DIM, 1);
    // layer 2
    hipMemsetAsync(aggS, 0, (size_t)2 * N_NODES * F_DIM * sizeof(float), stream);
    scatter_kernel<<<gScatter, 256, 0, stream>>>(h1, row, col, ew, aggS, aggD);
    gemm_dir_kernel<<<gGemmH, 256, 0, stream>>>(aggS, aggD, w_s2, w_d2, b_s2, b_d2,
                                                h2, N_NODES, F_DIM, 1);
    // layer 3 (output C=40, no relu)
    hipMemsetAsync(aggS, 0, (size_t)2 * N_NODES * F_DIM * sizeof(float), stream);
    scatter_kernel<<<gScatter, 256, 0, stream>>>(h2, row, col, ew, aggS, aggD);
    gemm_dir_kernel<<<gGemmC, 256, 0, stream>>>(aggS, aggD, w_s3, w_d3, b_s3, b_d3,
                                                out, N_NODES, N_CLS, 0);

    logsoftmax_kernel<<<(N_NODES + 7) / 8, 256, 0, stream>>>(out, N_NODES, N_CLS);
}